// FunctionToPatchInterpolationLayer_38001870635396
// MI455X (gfx1250) — compile-verified
//
#include <hip/hip_runtime.h>

// Problem constants (from reference)
#define N_PATCHES 5120
#define PTS 16
#define N_POINTS (N_PATCHES * PTS)   // 81920
#define N_DOF 163842
#define NFUNC 8
#define NB 16
#define PATCHES_PER_BLOCK 2
#define STAGE_ELEMS (PATCHES_PER_BLOCK * PTS * 3)   // 96 idx + 96 w

// ---- CDNA5 async global->LDS path (guarded; falls back to plain loads) ----
#if defined(__has_builtin)
#  if __has_builtin(__builtin_amdgcn_global_load_async_to_lds_b32) && \
      __has_builtin(__builtin_amdgcn_s_wait_asynccnt)
#    define HAVE_ASYNC_LDS 1
#  endif
#endif
#ifndef HAVE_ASYNC_LDS
#  define HAVE_ASYNC_LDS 0
#endif

// Builtin wants (__device__ int*) global src and (__shared__ int*) LDS dst
typedef __attribute__((address_space(1))) int* gip;
typedef __attribute__((address_space(3))) int* lip;

__global__ __launch_bounds__(256)
void interp_gather_kernel(const float* __restrict__ x,
                          const int*   __restrict__ interp_idx,
                          const float* __restrict__ interp_w,
                          float*       __restrict__ out)
{
    __shared__ int   sIdx[STAGE_ELEMS];
    __shared__ float sW[STAGE_ELEMS];

    const int tid        = threadIdx.x;
    const int blockPatch = blockIdx.x * PATCHES_PER_BLOCK;
    const int pointBase  = blockPatch * PTS;           // first point handled by block

    // ---- Stage idx/w for this block's 32 points into LDS ----
#if HAVE_ASYNC_LDS
    if (tid < STAGE_ELEMS) {
        __builtin_amdgcn_global_load_async_to_lds_b32(
            (gip)(interp_idx + (size_t)pointBase * 3 + tid),
            (lip)&sIdx[tid], 0, 0);
        __builtin_amdgcn_global_load_async_to_lds_b32(
            (gip)(interp_w + (size_t)pointBase * 3 + tid),
            (lip)&sW[tid], 0, 0);
        __builtin_amdgcn_s_wait_asynccnt(0);
    }
    __syncthreads();
#else
    if (tid < STAGE_ELEMS) {
        sIdx[tid] = interp_idx[(size_t)pointBase * 3 + tid];
        sW[tid]   = interp_w[(size_t)pointBase * 3 + tid];
    }
    __syncthreads();
#endif

    // ---- Thread decomposition: tid = (patchHalf<<7) | (f<<4) | pt ----
    const int pt        = tid & 15;
    const int f         = (tid >> 4) & 7;
    const int patchHalf = tid >> 7;                    // 0..1
    const int patch     = blockPatch + patchHalf;
    const int pLocal    = patchHalf * PTS + pt;        // 0..31 within block

    const int   i0 = sIdx[pLocal * 3 + 0];
    const int   i1 = sIdx[pLocal * 3 + 1];
    const int   i2 = sIdx[pLocal * 3 + 2];
    const float w0 = sW[pLocal * 3 + 0];
    const float w1 = sW[pLocal * 3 + 1];
    const float w2 = sW[pLocal * 3 + 2];

    // x[b][f][dof] : stride per b = NFUNC*N_DOF; per f = N_DOF
    const float* xf = x + (size_t)f * N_DOF;
    const size_t xStrideB = (size_t)NFUNC * N_DOF;

    // out[b][patch][f][pt] : contiguous in (f,pt) per (b,patch)
    float* op = out + (((size_t)patch * NFUNC + f) * PTS + pt);
    const size_t oStrideB = (size_t)N_PATCHES * NFUNC * PTS;

#pragma unroll
    for (int b = 0; b < NB; ++b) {
        const float* xb = xf + (size_t)b * xStrideB;
        // 3 random 4B gathers (L2-resident x) + FMA tree
        const float v = fmaf(w0, xb[i0], fmaf(w1, xb[i1], w2 * xb[i2]));
        // Output is write-once: stream it (non-temporal) to keep L2 for x
        __builtin_nontemporal_store(v, op + (size_t)b * oStrideB);
    }
}

extern "C" void kernel_launch(void* const* d_in, const int* in_sizes, int n_in,
                              void* d_out, int out_size, void* d_ws, size_t ws_size,
                              hipStream_t stream)
{
    const float* x          = (const float*)d_in[0];   // (16, 8, 163842) f32
    const int*   interp_idx = (const int*)d_in[1];     // (81920, 3) i32
    const float* interp_w   = (const float*)d_in[2];   // (81920, 3) f32
    float*       out        = (float*)d_out;           // (16, 5120, 8, 16) f32

    (void)in_sizes; (void)n_in; (void)out_size; (void)d_ws; (void)ws_size;

    const dim3 grid(N_PATCHES / PATCHES_PER_BLOCK);    // 2560 blocks
    const dim3 block(256);                             // 8 wave32 per block
    hipLaunchKernelGGL(interp_gather_kernel, grid, block, 0, stream,
                       x, interp_idx, interp_w, out);
}